// Similarity_69277822485047
// MI455X (gfx1250) — compile-verified
//
#include <hip/hip_runtime.h>
#include <stdint.h>

// Row-wise dot product: out[n] = sum_d x[n,d] * y[n,d]
// N = 16384 rows, D = 1024 cols, fp32. HBM-bandwidth bound (0.25 FLOP/byte,
// ~5.8us floor at 23.3 TB/s). Primary path: CDNA5 async global->LDS mover
// (GLOBAL_LOAD_ASYNC_TO_LDS_B128 + s_wait_asynccnt) with a per-wave
// double-buffered row pipeline. Fallback path: direct b128 loads to VGPRs.

#define DIMS 1024
#define WAVE 32
#define VEC_ITERS (DIMS / 4 / WAVE)   // 8 float4 (512B) transactions per row per input

// ---------------- wave32 butterfly reduction via ds_swizzle ----------------
#if __has_builtin(__builtin_amdgcn_ds_swizzle)
#define SWIZ_XOR(v, K) \
    __int_as_float(__builtin_amdgcn_ds_swizzle(__float_as_int(v), (((K) << 10) | 0x1F)))
#else
#define SWIZ_XOR(v, K) __shfl_xor((v), (K), WAVE)
#endif

__device__ __forceinline__ float wave_reduce(float v) {
    v += SWIZ_XOR(v, 16);
    v += SWIZ_XOR(v, 8);
    v += SWIZ_XOR(v, 4);
    v += SWIZ_XOR(v, 2);
    v += SWIZ_XOR(v, 1);
    return v;
}

// ---------------- primary kernel: async global->LDS pipelined ---------------
#if __has_builtin(__builtin_amdgcn_global_load_async_to_lds_b128)
#define HAVE_ASYNC_LDS 1
#else
#define HAVE_ASYNC_LDS 0
#endif

#if HAVE_ASYNC_LDS

#if __has_builtin(__builtin_amdgcn_s_wait_asynccnt)
#define WAIT_ASYNC(imm) __builtin_amdgcn_s_wait_asynccnt(imm)
#else
#define WAIT_ASYNC(imm) asm volatile("s_wait_asynccnt " #imm ::: "memory")
#endif

#define AW 4                     // waves per block
#define ROWS_PER_WAVE 4          // rows streamed per wave (software pipeline)
#define ATHREADS (AW * WAVE)
// per wave: double buffer of (x row + y row) = 2 * 2 * 4KB = 16KB -> 64KB/block.
// 1024 blocks / 4096 waves; steady state 16 async ops x 512B = 8KB in flight
// per wave -> ~32MB device-wide, above the ~19MB bandwidth-delay product.
#define SMEM_FLOATS (AW * 2 * 2 * DIMS)

// Builtin parameter type (from toolchain diagnostic): pointer to
// 'int __attribute__((vector_size(16)))' in AS1 (global) / AS3 (LDS).
typedef int v4i __attribute__((vector_size(16)));
typedef __attribute__((address_space(1))) v4i* g_v4i_ptr;
typedef __attribute__((address_space(3))) v4i* l_v4i_ptr;

#define AS_GLOBAL(p) ((g_v4i_ptr)(uintptr_t)(p))
#define AS_LDS(p)    ((l_v4i_ptr)(uintptr_t)(p))

// Issue one row (x + y) as 16 async 512B wave transactions into LDS.
__device__ __forceinline__ void issue_row_async(const float* gx, const float* gy,
                                                float* lx, float* ly, int lane) {
#pragma unroll
    for (int i = 0; i < VEC_ITERS; ++i) {
        const int off = (lane + WAVE * i) * 4;  // float index, 16B per lane
        __builtin_amdgcn_global_load_async_to_lds_b128(
            AS_GLOBAL(gx + off), AS_LDS(lx + off), 0, 0);
        __builtin_amdgcn_global_load_async_to_lds_b128(
            AS_GLOBAL(gy + off), AS_LDS(ly + off), 0, 0);
    }
}

__global__ void __launch_bounds__(ATHREADS)
rowdot_async(const float* __restrict__ x, const float* __restrict__ y,
             float* __restrict__ out, int n_full_rows) {
    __shared__ float smem[SMEM_FLOATS];
    const int lane  = threadIdx.x & (WAVE - 1);
    const int wave  = threadIdx.x >> 5;
    const int gwave = blockIdx.x * AW + wave;
    const int row0  = gwave * ROWS_PER_WAVE;
    if (row0 >= n_full_rows) return;

    float* wbase = &smem[wave * (2 * 2 * DIMS)];  // [buf][x|y][DIMS]

    // Prime the pipeline with row0 into buffer 0.
    issue_row_async(x + (size_t)row0 * DIMS, y + (size_t)row0 * DIMS,
                    wbase, wbase + DIMS, lane);

#pragma unroll 1
    for (int r = 0; r < ROWS_PER_WAVE; ++r) {
        const int b = r & 1;
        if (r + 1 < ROWS_PER_WAVE) {
            // Prefetch next row into the other buffer, then wait until only
            // those 16 new async ops remain outstanding -> current row landed
            // (async loads complete in order).
            float* nb = wbase + ((b ^ 1) * 2 * DIMS);
            const size_t nrow = (size_t)(row0 + r + 1);
            issue_row_async(x + nrow * DIMS, y + nrow * DIMS, nb, nb + DIMS, lane);
            WAIT_ASYNC(16);
        } else {
            WAIT_ASYNC(0);
        }

        const float* lx = wbase + (b * 2 * DIMS);
        const float* ly = lx + DIMS;
        float4 acc = make_float4(0.f, 0.f, 0.f, 0.f);
#pragma unroll
        for (int i = 0; i < VEC_ITERS; ++i) {
            const float4 a  = reinterpret_cast<const float4*>(lx)[lane + WAVE * i];
            const float4 bb = reinterpret_cast<const float4*>(ly)[lane + WAVE * i];
            acc.x = fmaf(a.x, bb.x, acc.x);
            acc.y = fmaf(a.y, bb.y, acc.y);
            acc.z = fmaf(a.z, bb.z, acc.z);
            acc.w = fmaf(a.w, bb.w, acc.w);
        }
        float v = (acc.x + acc.y) + (acc.z + acc.w);
        v = wave_reduce(v);
        if (lane == 0) out[row0 + r] = v;
    }
}
#endif  // HAVE_ASYNC_LDS

// ---------------- fallback / tail kernel: direct global b128 loads ----------
#define BWAVES 8
#define BTHREADS (BWAVES * WAVE)

__global__ void __launch_bounds__(BTHREADS)
rowdot_direct(const float* __restrict__ x, const float* __restrict__ y,
              float* __restrict__ out, int row_base, int n) {
    const int lane = threadIdx.x & (WAVE - 1);
    const int wave = threadIdx.x >> 5;
    const int row  = row_base + blockIdx.x * BWAVES + wave;
    if (row >= n) return;  // uniform across the wave

    const float4* __restrict__ px =
        reinterpret_cast<const float4*>(x + (size_t)row * DIMS);
    const float4* __restrict__ py =
        reinterpret_cast<const float4*>(y + (size_t)row * DIMS);

    float4 acc = make_float4(0.f, 0.f, 0.f, 0.f);
#pragma unroll
    for (int i = 0; i < VEC_ITERS; ++i) {
        const float4 a = px[lane + WAVE * i];
        const float4 b = py[lane + WAVE * i];
        acc.x = fmaf(a.x, b.x, acc.x);
        acc.y = fmaf(a.y, b.y, acc.y);
        acc.z = fmaf(a.z, b.z, acc.z);
        acc.w = fmaf(a.w, b.w, acc.w);
    }
    float v = (acc.x + acc.y) + (acc.z + acc.w);
    v = wave_reduce(v);
    if (lane == 0) out[row] = v;
}

// ---------------- launch ----------------------------------------------------
extern "C" void kernel_launch(void* const* d_in, const int* in_sizes, int n_in,
                              void* d_out, int out_size, void* d_ws, size_t ws_size,
                              hipStream_t stream) {
    (void)n_in; (void)d_ws; (void)ws_size; (void)in_sizes;

    const float* x = reinterpret_cast<const float*>(d_in[0]);
    const float* y = reinterpret_cast<const float*>(d_in[1]);
    float* out     = reinterpret_cast<float*>(d_out);
    const int n    = out_size;  // 16384 rows

#if HAVE_ASYNC_LDS
    const int rows_per_block = AW * ROWS_PER_WAVE;          // 16
    const int full_blocks    = n / rows_per_block;          // 1024 for N=16384
    const int bulk_rows      = full_blocks * rows_per_block;
    if (full_blocks > 0) {
        rowdot_async<<<full_blocks, ATHREADS, 0, stream>>>(x, y, out, bulk_rows);
    }
    const int rem = n - bulk_rows;
    if (rem > 0) {
        const int blocks = (rem + BWAVES - 1) / BWAVES;
        rowdot_direct<<<blocks, BTHREADS, 0, stream>>>(x, y, out, bulk_rows, n);
    }
#else
    const int blocks = (n + BWAVES - 1) / BWAVES;
    rowdot_direct<<<blocks, BTHREADS, 0, stream>>>(x, y, out, 0, n);
#endif
}